// ROIAlignBatch_17188459118614
// MI455X (gfx1250) — compile-verified
//
#include <hip/hip_runtime.h>

// ---------------- CDNA5 async global->LDS copy plumbing ----------------------
#if __has_builtin(__builtin_amdgcn_global_load_async_to_lds_b32)
#warning PROBE: using __builtin_amdgcn_global_load_async_to_lds_b32
typedef __attribute__((address_space(1))) int cdna5_gint;   // global (device) int
typedef __attribute__((address_space(3))) int cdna5_lint;   // LDS (shared) int
#define ASYNC_COPY_B32(gsrc, ldst)                                            \
    __builtin_amdgcn_global_load_async_to_lds_b32(                            \
        (cdna5_gint*)(gsrc), (cdna5_lint*)(ldst), 0, 0)
#else
#warning PROBE: async builtin NOT found - using inline asm global_load_async_to_lds_b32
#define ASYNC_COPY_B32(gsrc, ldst)                                            \
    asm volatile("global_load_async_to_lds_b32 %0, %1, off"                   \
                 :: "v"((unsigned)(unsigned long long)                        \
                        (__attribute__((address_space(3))) void*)(ldst)),     \
                    "v"((unsigned long long)(gsrc))                           \
                 : "memory")
#endif

#if __has_builtin(__builtin_amdgcn_s_wait_asynccnt)
#warning PROBE: using __builtin_amdgcn_s_wait_asynccnt
#define ASYNC_WAIT0() __builtin_amdgcn_s_wait_asynccnt(0)
#else
#warning PROBE: s_wait_asynccnt builtin NOT found - using inline asm
#define ASYNC_WAIT0() asm volatile("s_wait_asynccnt 0x0" ::: "memory")
#endif

#define POOL   7
#define NBIN   49                 // 7*7
#define C_     256
#define H_     200
#define W_     304
#define HW_    (H_ * W_)
#define N_     512                // ROIs per image
#define BUF_DW 10240              // 40KB LDS patch buffer (in dwords)
// Analytic worst-case patch: patchW <= 61, patchH <= 42 -> patchSz <= 2562 dwords,
// so BUF_DW always holds >= 3 channels.

// One workgroup per ROI. 256 threads = 8 wave32.
// Phase 1: build 49*4 sample table (patch-RELATIVE offsets + weights, 0.25 & validity folded).
// Stage loop: async bulk copy of the ROI patch for nch channels into LDS
//             (coalesced rows, ASYNCcnt-tracked), then bilinear gather from LDS.
// lanes <-> bins (lane owns bin=lane and, for lanes<17, bin=lane+32) -> coalesced stores.
__global__ __launch_bounds__(256)
void roialign_gfx1250_async_kernel(const float* __restrict__ feat,
                                   const float* __restrict__ rois,
                                   const int*   __restrict__ stride_p,
                                   float*       __restrict__ out)
{
    __shared__ float s_buf[BUF_DW];          // staged patches: [cc][patchH*patchW]
    __shared__ int   s_off[NBIN * 4 * 4];    // patch-relative corner offsets
    __shared__ float s_w  [NBIN * 4 * 4];    // folded bilinear weights

    const int blk = blockIdx.x;              // b*N_ + n
    const int tid = threadIdx.x;
    const int wave = tid >> 5;
    const int lane = tid & 31;

    // ---- uniform ROI geometry (all threads; scalarizes) --------------------
    const float scale = 1.0f / (float)stride_p[0];
    const float* box  = rois + (size_t)blk * 4;
    const float bx1 = box[0] * scale;
    const float by1 = box[1] * scale;
    const float bx2 = box[2] * scale;
    const float by2 = box[3] * scale;
    const float bw = fmaxf(bx2 - bx1, 1.0f) * (1.0f / (float)POOL);
    const float bh = fmaxf(by2 - by1, 1.0f) * (1.0f / (float)POOL);

    // Patch bounds from first/last sample coords (bit-identical expressions to
    // the table build below; clamp/floor are monotone, so these bracket all corners).
    const float xf = fminf(fmaxf(bx1 + (0.0f + 0.25f) * bw, 0.0f), (float)(W_ - 1));
    const float xl = fminf(fmaxf(bx1 + (6.0f + 0.75f) * bw, 0.0f), (float)(W_ - 1));
    const float yf = fminf(fmaxf(by1 + (0.0f + 0.25f) * bh, 0.0f), (float)(H_ - 1));
    const float yl = fminf(fmaxf(by1 + (6.0f + 0.75f) * bh, 0.0f), (float)(H_ - 1));
    const int x_lo = (int)floorf(xf);
    const int y_lo = (int)floorf(yf);
    const int x_hi = min((int)floorf(xl) + 1, W_ - 1);
    const int y_hi = min((int)floorf(yl) + 1, H_ - 1);
    const int patchW  = x_hi - x_lo + 1;     // <= 61
    const int patchH  = y_hi - y_lo + 1;     // <= 42
    const int patchSz = patchW * patchH;     // <= 2562 dwords

    int nch = BUF_DW / patchSz;              // channels per stage (>= 3)
    if (nch >= 8)  nch &= ~7;                // multiple of 8 -> balanced waves
    if (nch > 32)  nch = 32;

    // ---- Phase 1: sample table (196 threads) -------------------------------
    if (tid < NBIN * 4) {
        const int bin = tid >> 2;            // 0..48
        const int s   = tid & 3;             // sy = s>>1, sx = s&1
        const int ph  = bin / POOL;
        const int pw  = bin % POOL;
        const float offy = ((float)(s >> 1) + 0.5f) * 0.5f;
        const float offx = ((float)(s &  1) + 0.5f) * 0.5f;

        float y = by1 + ((float)ph + offy) * bh;
        float x = bx1 + ((float)pw + offx) * bw;
        const bool valid = (y > -1.0f) && (y < (float)H_) && (x > -1.0f) && (x < (float)W_);
        y = fminf(fmaxf(y, 0.0f), (float)(H_ - 1));
        x = fminf(fmaxf(x, 0.0f), (float)(W_ - 1));

        const int y0  = (int)floorf(y);
        const int x0  = (int)floorf(x);
        const int y1i = min(y0 + 1, H_ - 1);
        const int x1i = min(x0 + 1, W_ - 1);
        const float ly = y - (float)y0, lx = x - (float)x0;
        const float hy = 1.0f - ly,     hx = 1.0f - lx;
        const float v = valid ? 0.25f : 0.0f;   // fold sample-mean + validity

        const int r0 = y0  - y_lo, r1 = y1i - y_lo;
        const int q0 = x0  - x_lo, q1 = x1i - x_lo;
        const int base = tid * 4;
        s_off[base + 0] = r0 * patchW + q0;
        s_off[base + 1] = r0 * patchW + q1;
        s_off[base + 2] = r1 * patchW + q0;
        s_off[base + 3] = r1 * patchW + q1;
        s_w  [base + 0] = hy * hx * v;
        s_w  [base + 1] = hy * lx * v;
        s_w  [base + 2] = ly * hx * v;
        s_w  [base + 3] = ly * lx * v;
    }
    __syncthreads();

    // ---- register-cache the table (lane owns bin0=lane, bin1=lane+32) -----
    const int bin0 = lane;
    const int bin1 = lane + 32;
    const bool has2 = (bin1 < NBIN);
    int o0[16]; float w0[16];
    int o1[16]; float w1[16];
#pragma unroll
    for (int k = 0; k < 16; ++k) {
        o0[k] = s_off[bin0 * 16 + k];
        w0[k] = s_w  [bin0 * 16 + k];
    }
    if (has2) {
#pragma unroll
        for (int k = 0; k < 16; ++k) {
            o1[k] = s_off[bin1 * 16 + k];
            w1[k] = s_w  [bin1 * 16 + k];
        }
    }

    const float* featb = feat + (size_t)(blk / N_) * (size_t)C_ * HW_;
    float*       outb  = out  + (size_t)blk * (size_t)(C_ * NBIN);
    const float* patch0 = featb + (size_t)y_lo * W_ + x_lo;

    // ---- stage loop --------------------------------------------------------
    int stg = 0;
    for (int c0 = 0; c0 < C_; c0 += nch, ++stg) {
        const int m = min(nch, C_ - c0);

        // async bulk copy: m channel-patches, coalesced rows -> LDS
        for (int cc = 0; cc < m; ++cc) {
            const float* src = patch0 + (size_t)(c0 + cc) * HW_;
            const int ldsc = cc * patchSz;
            for (int r = wave; r < patchH; r += 8) {
                const float* rowp = src + r * W_;
                const int ldsr = ldsc + r * patchW;
                for (int col = lane; col < patchW; col += 32) {
                    ASYNC_COPY_B32(rowp + col, &s_buf[ldsr + col]);
                }
            }
        }
        ASYNC_WAIT0();          // my async writes have landed in LDS
        __syncthreads();        // everyone's have

        // bilinear gather from LDS; rotate channel ownership to balance m%8!=0
        for (int cc = ((wave + stg) & 7); cc < m; cc += 8) {
            const int c = c0 + cc;
            const int base = cc * patchSz;
            float a0 = 0.0f;
#pragma unroll
            for (int k = 0; k < 16; ++k) a0 = fmaf(w0[k], s_buf[base + o0[k]], a0);
            outb[c * NBIN + bin0] = a0;
            if (has2) {
                float a1 = 0.0f;
#pragma unroll
                for (int k = 0; k < 16; ++k) a1 = fmaf(w1[k], s_buf[base + o1[k]], a1);
                outb[c * NBIN + bin1] = a1;
            }
        }
        __syncthreads();        // all reads done before next stage overwrites s_buf
    }
}

extern "C" void kernel_launch(void* const* d_in, const int* in_sizes, int n_in,
                              void* d_out, int out_size, void* d_ws, size_t ws_size,
                              hipStream_t stream) {
    (void)n_in; (void)out_size; (void)d_ws; (void)ws_size;
    const float* feat   = (const float*)d_in[0];   // (4,256,200,304) f32
    const float* rois   = (const float*)d_in[1];   // (4,512,4) f32
    const int*   stride = (const int*)d_in[2];     // scalar int (=4)
    float*       out    = (float*)d_out;           // (4,512,256,7,7) f32

    const int n_rois = in_sizes[1] / 4;            // 2048 workgroups
    hipLaunchKernelGGL(roialign_gfx1250_async_kernel,
                       dim3(n_rois), dim3(256), 0, stream,
                       feat, rois, stride, out);
}